// distance_35570919146090
// MI455X (gfx1250) — compile-verified
//
#include <hip/hip_runtime.h>
#include <math.h>

typedef __attribute__((ext_vector_type(2))) float v2f;
typedef __attribute__((ext_vector_type(8))) float v8f;

#define DF 64

__global__ void zero_f32_kernel(float* __restrict__ p, int n) {
  int i = blockIdx.x * blockDim.x + threadIdx.x;
  if (i < n) p[i] = 0.0f;
}

__device__ __forceinline__ float pick8(v8f c, int i) {
  switch (i & 7) {
    case 0: return c[0];
    case 1: return c[1];
    case 2: return c[2];
    case 3: return c[3];
    case 4: return c[4];
    case 5: return c[5];
    case 6: return c[6];
    default: return c[7];
  }
}

// One wave32 per 16-edge tile. Dot products via V_WMMA_F32_16X16X4_F32
// (diagonal of A[src rows] x B[dst cols]); then e = exp(-0.01*|dot|),
// ex = exp(e), atomic segment-sum into seg_sum[dst].
__global__ void edge_dot_exp_kernel(const float* __restrict__ feats,
                                    const int* __restrict__ src,
                                    const int* __restrict__ dst,
                                    float* __restrict__ ex_out,
                                    float* __restrict__ seg_sum,
                                    int n_edges) {
  const int lane = threadIdx.x & 31;
  const int waveInBlock = threadIdx.x >> 5;
  const int wavesPerBlock = blockDim.x >> 5;
  const long long tile = (long long)blockIdx.x * wavesPerBlock + waveInBlock;
  const long long base = tile * 16;
  if (base >= (long long)n_edges) return;  // wave-uniform exit

  const int m = lane & 15;        // row within tile (A: M, B: N)
  const int half = lane >> 4;     // K-half selector

  if (base + 16 <= (long long)n_edges) {
    // ---- full tile: WMMA path (whole wave active, EXEC all ones) ----
    const long long e = base + (long long)m;
    const int si = src[e];
    const int di = dst[e];
    // A 16x4 f32 layout: lane(M=m), VGPR0/1 = K = 2*half, 2*half+1
    // B 4x16 f32 layout: lane(N=m), VGPR0/1 = K = 2*half, 2*half+1
    const float* arow = feats + (long long)si * DF + 2 * half;
    const float* brow = feats + (long long)di * DF + 2 * half;

    v8f c = {};
#pragma unroll
    for (int s = 0; s < DF / 4; ++s) {
      v2f a = *(const v2f*)(arow + 4 * s);
      v2f b = *(const v2f*)(brow + 4 * s);
      // 8 args: (neg_a, A, neg_b, B, c_mod, C, reuse_a, reuse_b)
      c = __builtin_amdgcn_wmma_f32_16x16x4_f32(
          false, a, false, b, (short)0, c, false, false);
    }

    // Diagonal D[m][m]: m<8 -> VGPR m @ lane m ; m>=8 -> VGPR m-8 @ lane m+8
    int myM = -1;
    float d = 0.0f;
    if (lane < 8) {
      myM = lane;
      d = pick8(c, lane);
    } else if (lane >= 16 && lane < 24) {
      myM = lane - 8;
      d = pick8(c, lane - 16);
    }
    if (myM >= 0) {
      const long long ei = base + (long long)myM;
      const float ev = expf(-0.01f * fabsf(d));
      const float exv = expf(ev);  // shift-free softmax numerator (e in (0,1])
      ex_out[ei] = exv;
      atomicAdd(&seg_sum[dst[ei]], exv);
    }
  } else {
    // ---- remainder tile: scalar per-lane dot (lanes 0..rem-1) ----
    const long long e = base + (long long)lane;
    if (lane < 16 && e < (long long)n_edges) {
      const float* ar = feats + (long long)src[e] * DF;
      const float* br = feats + (long long)dst[e] * DF;
      float d = 0.0f;
#pragma unroll 8
      for (int k = 0; k < DF; ++k) d += ar[k] * br[k];
      const float ev = expf(-0.01f * fabsf(d));
      const float exv = expf(ev);
      ex_out[e] = exv;
      atomicAdd(&seg_sum[dst[e]], exv);
    }
  }
}

__global__ void normalize_kernel(const float* __restrict__ ex,
                                 const int* __restrict__ dst,
                                 const float* __restrict__ seg_sum,
                                 float* __restrict__ out, int n_edges) {
  int i = blockIdx.x * blockDim.x + threadIdx.x;
  if (i < n_edges) out[i] = ex[i] / seg_sum[dst[i]];
}

extern "C" void kernel_launch(void* const* d_in, const int* in_sizes, int n_in,
                              void* d_out, int out_size, void* d_ws, size_t ws_size,
                              hipStream_t stream) {
  const float* feats = (const float*)d_in[0];
  const int* src = (const int*)d_in[1];
  const int* dst = (const int*)d_in[2];

  const int n_nodes = in_sizes[0] / DF;
  const int n_edges = in_sizes[1];

  float* seg_sum = (float*)d_ws;          // n_nodes floats
  float* ex = seg_sum + n_nodes;          // n_edges floats

  // Pass 0: zero the segment sums (workspace is poisoned, not zeroed).
  zero_f32_kernel<<<(n_nodes + 255) / 256, 256, 0, stream>>>(seg_sum, n_nodes);

  // Pass 1: WMMA dot products + exp + atomic segment sums.
  const int tiles = (n_edges + 15) / 16;
  const int wavesPerBlock = 8;  // 256 threads = 8 wave32
  const int blocks = (tiles + wavesPerBlock - 1) / wavesPerBlock;
  edge_dot_exp_kernel<<<blocks, 256, 0, stream>>>(feats, src, dst, ex, seg_sum,
                                                  n_edges);

  // Pass 2: normalize per edge.
  normalize_kernel<<<(n_edges + 255) / 256, 256, 0, stream>>>(
      ex, dst, seg_sum, (float*)d_out, n_edges);
}